// ECPEC_37512244363404
// MI455X (gfx1250) — compile-verified
//
#include <hip/hip_runtime.h>
#include <math.h>

typedef float v2f __attribute__((ext_vector_type(2)));
typedef float v8f __attribute__((ext_vector_type(8)));

// ---------------------------------------------------------------------------
// wave32 WMMA f32 16x16x4 wrapper
// ---------------------------------------------------------------------------
__device__ __forceinline__ v8f wmma_f32(v2f a, v2f b, v8f c) {
  return __builtin_amdgcn_wmma_f32_16x16x4_f32(false, a, false, b, (short)0, c,
                                               false, false);
}

// ---------------------------------------------------------------------------
// Generic GEMM: C[M,N] = act(A[M,K] @ B[K,N] + bias[N]); K % 4 == 0,
// M,N % 16 == 0. One wave per 16x16 C tile. grid = (N/16, M/16), block = 32.
// act: 0 = none, 1 = relu.
// ---------------------------------------------------------------------------
__global__ __launch_bounds__(32) void gemm_wmma_f32_kernel(
    const float* __restrict__ A, const float* __restrict__ B,
    const float* __restrict__ bias, float* __restrict__ C, int M, int N, int K,
    int act) {
  const int nt = blockIdx.x, mt = blockIdx.y;
  const int lane = threadIdx.x;
  const int msub = lane & 15;
  const int ncol = nt * 16 + (lane & 15);
  const int koff = (lane >> 4) * 2;  // ISA A layout: lanes 16-31 hold K+2,K+3
  const float* arow = A + (mt * 16 + msub) * K;
  v8f acc = {};
  for (int k0 = 0; k0 < K; k0 += 4) {
    const int ka = k0 + koff;
    v2f a = *(const v2f*)(arow + ka);  // 8B aligned: K%4==0, ka even
    v2f b;
    b.x = B[ka * N + ncol];
    b.y = B[(ka + 1) * N + ncol];
    acc = wmma_f32(a, b, acc);
  }
  // C/D layout: vgpr v -> M = v (lanes 0-15) / v+8 (lanes 16-31)
  const int mbase = mt * 16 + (lane >> 4) * 8;
  const float bv = bias ? bias[ncol] : 0.0f;
#pragma unroll
  for (int v = 0; v < 8; ++v) {
    float h = acc[v] + bv;
    if (act) h = fmaxf(h, 0.0f);
    C[(mbase + v) * N + ncol] = h;
  }
}

// ---------------------------------------------------------------------------
__global__ __launch_bounds__(256) void zero_f32_kernel(float* p, int n) {
  int t = blockIdx.x * 256 + threadIdx.x;
  if (t < n) p[t] = 0.0f;
}

// ---------------------------------------------------------------------------
// Squared row norms of emo (rows 0..511) and cau (rows 512..1023).
// One wave per row, grid = 128 blocks of 256.
// ---------------------------------------------------------------------------
__global__ __launch_bounds__(256) void sqnorm_kernel(
    const float* __restrict__ emo, const float* __restrict__ cau,
    float* __restrict__ sqne, float* __restrict__ sqnc) {
  const int row = blockIdx.x * 8 + (threadIdx.x >> 5);  // 0..1023
  const int lane = threadIdx.x & 31;
  const float* src = (row < 512) ? (emo + row * 256) : (cau + (row - 512) * 256);
  float acc = 0.0f;
  for (int k = lane; k < 256; k += 32) {
    const float v = src[k];
    acc += v * v;
  }
  for (int o = 16; o > 0; o >>= 1) acc += __shfl_xor(acc, o, 32);
  if (lane == 0) {
    if (row < 512) sqne[row] = acc;
    else sqnc[row - 512] = acc;
  }
}

// ---------------------------------------------------------------------------
// GRU gate update for step t. GX: (512,768) precomputed x-gates (row c*8+t),
// GH: (64,768) h-gates for this step. H: (64,256) updated in place.
// ---------------------------------------------------------------------------
__global__ __launch_bounds__(256) void gru_gate_kernel(
    const float* __restrict__ GX, const float* __restrict__ GH,
    float* __restrict__ H, int t) {
  const int tid = blockIdx.x * 256 + threadIdx.x;  // 64*256
  const int c = tid >> 8, d = tid & 255;
  const float* gx = GX + (c * 8 + t) * 768;
  const float* gh = GH + c * 768;
  const float r = 1.0f / (1.0f + expf(-(gx[d] + gh[d])));
  const float z = 1.0f / (1.0f + expf(-(gx[256 + d] + gh[256 + d])));
  const float n = tanhf(gx[512 + d] + r * gh[512 + d]);
  const float h = H[c * 256 + d];
  H[c * 256 + d] = (1.0f - z) * n + z * h;
}

// ---------------------------------------------------------------------------
// phase2: per (i,c) pair: hidden_k = sigmoid(P[i,k]+Q[c,k]) (P has W_b folded),
// z = hidden @ Wo_W + Wo_b, out = log_softmax(z). One wave per pair.
// ---------------------------------------------------------------------------
__global__ __launch_bounds__(256) void phase2_kernel(
    const float* __restrict__ P, const float* __restrict__ Q,
    const float* __restrict__ WoW, const float* __restrict__ Wob,
    float* __restrict__ out) {
  const int pair = blockIdx.x * 8 + (threadIdx.x >> 5);  // i*64 + c
  const int lane = threadIdx.x & 31;
  const float* prow = P + (pair >> 6) * 256;
  const float* qrow = Q + (pair & 63) * 256;
  float a0 = 0.0f, a1 = 0.0f;
  for (int k = lane; k < 256; k += 32) {
    const float h = 1.0f / (1.0f + expf(-(prow[k] + qrow[k])));
    a0 += h * WoW[2 * k];
    a1 += h * WoW[2 * k + 1];
  }
  for (int o = 16; o > 0; o >>= 1) {
    a0 += __shfl_xor(a0, o, 32);
    a1 += __shfl_xor(a1, o, 32);
  }
  if (lane == 0) {
    const float z0 = a0 + Wob[0], z1 = a1 + Wob[1];
    const float m = fmaxf(z0, z1);
    const float lse = m + logf(expf(z0 - m) + expf(z1 - m));
    out[2 * pair] = z0 - lse;
    out[2 * pair + 1] = z1 - lse;
  }
}

// ---------------------------------------------------------------------------
// Pair kernel (dominant 34 GFLOP stage).
// block b: i = b>>4, jb = (b&15)*32. 256 threads = 8 waves.
// GEMM computed transposed:  C[n,j] = sum_k W3d[k,n] * prodT[k,j]
//   A = W3d^T streamed from global (coalesced 64B rows, 4 b32/lane/K-step)
//   B = prodT[k][j] = emo_i[k]*cau[jb+j,k] staged once in LDS (stride 33)
// norms: ||e-c||^2 = sqne[i] + sqnc[j] - 2 * colsum(prodT)
//   h3[j,n] = relu(C + A3[i,n] + B3[j,n] + norm*w3c[n])   (A3 holds b3)
//   p[i,j,:] = log_softmax(h3 @ cls_W + cls_b);  h3 lives only in LDS.
// ---------------------------------------------------------------------------
#define TSTR 33   // prodT row stride (k-major)
#define HSTR 257  // h3 row stride (j-major)
__global__ __launch_bounds__(256) void pair_kernel(
    const float* __restrict__ cau, const float* __restrict__ emo,
    const float* __restrict__ A3, const float* __restrict__ B3,
    const float* __restrict__ W3,  // (769,256): w3c = row 512, W3d = rows 513+
    const float* __restrict__ clsW, const float* __restrict__ clsb,
    const float* __restrict__ sqne, const float* __restrict__ sqnc,
    float* __restrict__ pout) {
  __shared__ float s_emo[256];
  __shared__ float s_a3[256];
  __shared__ float s_w3c[256];
  __shared__ float s_norm[32];
  __shared__ float s_buf[256 * TSTR];  // prodT(k,j)=k*33+j ; h3(j,n)=j*257+n

  const int i = blockIdx.x >> 4;
  const int jb = (blockIdx.x & 15) * 32;
  const int t = threadIdx.x;
  const int wave = t >> 5, lane = t & 31;
  const float* w3c = W3 + 512 * 256;
  const float* W3d = W3 + 513 * 256;

  // ---- stage per-i vectors
  s_emo[t] = emo[i * 256 + t];
  s_a3[t] = A3[i * 256 + t];
  s_w3c[t] = w3c[t];
  __syncthreads();

  // ---- build transposed product tile (coalesced cau reads)
  for (int x = t; x < 32 * 256; x += 256) {
    const int j = x >> 8, k = x & 255;
    s_buf[k * TSTR + j] = cau[(jb + j) * 256 + k] * s_emo[k];
  }
  __syncthreads();

  // ---- pair norms from prodT column sums (wave w owns j = 4w..4w+3)
  {
    const float se = sqne[i];
    for (int rr = 0; rr < 4; ++rr) {
      const int j = wave * 4 + rr;
      float acc = 0.0f;
      for (int k = lane; k < 256; k += 32) acc += s_buf[k * TSTR + j];
      for (int o = 16; o > 0; o >>= 1) acc += __shfl_xor(acc, o, 32);
      if (lane == 0)
        s_norm[j] = sqrtf(fmaxf(se + sqnc[jb + j] - 2.0f * acc, 0.0f));
    }
  }

  // ---- WMMA: wave w owns n-tiles {2w, 2w+1} x j-tiles {0,1}
  const int n0 = wave * 32;
  const int koff = (lane >> 4) * 2;  // A/B frag: lanes 16-31 hold K+2,K+3
  const int sub = lane & 15;

  v8f acc00 = {}, acc01 = {}, acc10 = {}, acc11 = {};
  for (int kp = 0; kp < 256; kp += 16) {
    if (kp + 16 < 256)  // warm next W3d panel (global_prefetch_b8)
      __builtin_prefetch(W3d + (kp + 16 + koff) * 256 + n0 + sub, 0, 3);
#pragma unroll
    for (int kk = 0; kk < 16; kk += 4) {
      const int ka = kp + kk + koff;
      const float* wr0 = W3d + ka * 256;
      const float* wr1 = wr0 + 256;
      v2f a0, a1, b0, b1;
      a0.x = wr0[n0 + sub];       // coalesced: 16 consecutive floats/row
      a0.y = wr1[n0 + sub];
      a1.x = wr0[n0 + 16 + sub];
      a1.y = wr1[n0 + 16 + sub];
      b0.x = s_buf[ka * TSTR + sub];
      b0.y = s_buf[(ka + 1) * TSTR + sub];
      b1.x = s_buf[ka * TSTR + 16 + sub];
      b1.y = s_buf[(ka + 1) * TSTR + 16 + sub];
      acc00 = wmma_f32(a0, b0, acc00);
      acc01 = wmma_f32(a0, b1, acc01);
      acc10 = wmma_f32(a1, b0, acc10);
      acc11 = wmma_f32(a1, b1, acc11);
    }
  }
  __syncthreads();  // all waves done reading prodT / writing norms

  // ---- epilogue: h3 into LDS (s_buf reused, j-major stride 257)
  // C layout: (vgpr v, lane L): n = ntile + v + (L>>4)*8, j = jtile + (L&15)
  {
    const int jB = sub;  // j offset within j-tile
#pragma unroll
    for (int tile = 0; tile < 4; ++tile) {
      const v8f* accp = (tile == 0)   ? &acc00
                        : (tile == 1) ? &acc01
                        : (tile == 2) ? &acc10
                                      : &acc11;
      const int nt = n0 + (tile >> 1) * 16 + ((lane >> 4) << 3);
      const int j = (tile & 1) * 16 + jB;
      const float nrm = s_norm[j];
      const float* b3row = B3 + (jb + j) * 256;
#pragma unroll
      for (int v = 0; v < 8; ++v) {
        const int n = nt + v;  // consecutive per v -> b128-mergeable B3 reads
        const float h = (*accp)[v] + s_a3[n] + b3row[n] + nrm * s_w3c[n];
        s_buf[j * HSTR + n] = fmaxf(h, 0.0f);
      }
    }
  }
  __syncthreads();

  // ---- fused cls projection + log_softmax (2 classes)
  for (int rr = 0; rr < 4; ++rr) {
    const int j = wave * 4 + rr;
    float a0 = 0.0f, a1 = 0.0f;
    for (int k = lane; k < 256; k += 32) {
      const float h = s_buf[j * HSTR + k];
      a0 += h * clsW[2 * k];
      a1 += h * clsW[2 * k + 1];
    }
    for (int o = 16; o > 0; o >>= 1) {
      a0 += __shfl_xor(a0, o, 32);
      a1 += __shfl_xor(a1, o, 32);
    }
    if (lane == 0) {
      const float z0 = a0 + clsb[0], z1 = a1 + clsb[1];
      const float m = fmaxf(z0, z1);
      const float lse = m + logf(expf(z0 - m) + expf(z1 - m));
      const long idx = (long)(i * 512 + jb + j) * 2;
      pout[idx] = z0 - lse;
      pout[idx + 1] = z1 - lse;
    }
  }
}

// ---------------------------------------------------------------------------
__global__ __launch_bounds__(256) void lcau_kernel(const int* __restrict__ l3,
                                                   float* __restrict__ out) {
  const int tid = blockIdx.x * 256 + threadIdx.x;  // 512*512
  const int i = tid >> 9, j = tid & 511;
  const int* r = l3 + i * 3;
  out[tid] = (j == r[0] || j == r[1] || j == r[2]) ? 1.0f : 0.0f;
}

// ---------------------------------------------------------------------------
extern "C" void kernel_launch(void* const* d_in, const int* in_sizes, int n_in,
                              void* d_out, int out_size, void* d_ws,
                              size_t ws_size, hipStream_t stream) {
  const float* text_emo = (const float*)d_in[0];  // (512,1,256)
  const float* text_cau = (const float*)d_in[1];  // (512,1,256)
  const int* label3 = (const int*)d_in[3];        // (512,1,3)
  const float* rep_W = (const float*)d_in[5];     // (256,256)
  const float* rep_b = (const float*)d_in[6];     // (256)
  const float* W_W = (const float*)d_in[7];       // (512,256)
  const float* W_b = (const float*)d_in[8];       // (256)
  const float* Wo_W = (const float*)d_in[9];      // (256,2)
  const float* Wo_b = (const float*)d_in[10];     // (2)
  const float* gru_Wih = (const float*)d_in[11];  // (256,768)
  const float* gru_Whh = (const float*)d_in[12];  // (256,768)
  const float* gru_bih = (const float*)d_in[13];  // (768)
  const float* gru_bhh = (const float*)d_in[14];  // (768)
  const float* W3_W = (const float*)d_in[15];     // (769,256)
  const float* W3_b = (const float*)d_in[16];     // (256)
  const float* cls_W = (const float*)d_in[17];    // (256,2)
  const float* cls_b = (const float*)d_in[18];    // (2)

  float* out = (float*)d_out;
  float* phase2 = out;         // 512*64*2
  float* p = out + 65536;      // 512*512*2
  float* lcau = out + 589824;  // 512*512

  // workspace (~4.4 MB of f32)
  float* w = (float*)d_ws;
  float* emo = w;             // 512*256
  float* cau = emo + 131072;  // 512*256
  float* GX = cau + 131072;   // 512*768
  float* H = GX + 393216;     // 64*256
  float* GH = H + 16384;      // 64*768
  float* Q = GH + 49152;      // 64*256
  float* P = Q + 16384;       // 512*256
  float* A3 = P + 131072;     // 512*256
  float* B3 = A3 + 131072;    // 512*256
  float* sqne = B3 + 131072;  // 512
  float* sqnc = sqne + 512;   // 512
  (void)in_sizes; (void)n_in; (void)out_size; (void)ws_size;

  // 1) rep projections: emo/cau = relu(x @ rep_W + rep_b)
  gemm_wmma_f32_kernel<<<dim3(16, 32), 32, 0, stream>>>(text_emo, rep_W, rep_b,
                                                        emo, 512, 256, 256, 1);
  gemm_wmma_f32_kernel<<<dim3(16, 32), 32, 0, stream>>>(text_cau, rep_W, rep_b,
                                                        cau, 512, 256, 256, 1);
  sqnorm_kernel<<<128, 256, 0, stream>>>(emo, cau, sqne, sqnc);

  // 2) GRU: precompute all x-gates in one GEMM, then 8 sequential h-steps
  gemm_wmma_f32_kernel<<<dim3(48, 32), 32, 0, stream>>>(cau, gru_Wih, gru_bih,
                                                        GX, 512, 768, 256, 0);
  zero_f32_kernel<<<64, 256, 0, stream>>>(H, 16384);
  for (int t = 0; t < 8; ++t) {
    gemm_wmma_f32_kernel<<<dim3(48, 4), 32, 0, stream>>>(H, gru_Whh, gru_bhh,
                                                         GH, 64, 768, 256, 0);
    gru_gate_kernel<<<64, 256, 0, stream>>>(GX, GH, H, t);
  }

  // 3) hidden decomposition: P = emo@W_W[:256]+W_b,  Q = ch_last@W_W[256:]
  gemm_wmma_f32_kernel<<<dim3(16, 32), 32, 0, stream>>>(emo, W_W, W_b, P, 512,
                                                        256, 256, 0);
  gemm_wmma_f32_kernel<<<dim3(16, 4), 32, 0, stream>>>(H, W_W + 256 * 256,
                                                       nullptr, Q, 64, 256, 256,
                                                       0);
  phase2_kernel<<<4096, 256, 0, stream>>>(P, Q, Wo_W, Wo_b, phase2);

  // 4) W3 decomposition precomputes: A3 = emo@W3a + b3, B3 = cau@W3b
  gemm_wmma_f32_kernel<<<dim3(16, 32), 32, 0, stream>>>(emo, W3_W, W3_b, A3,
                                                        512, 256, 256, 0);
  gemm_wmma_f32_kernel<<<dim3(16, 32), 32, 0, stream>>>(cau, W3_W + 256 * 256,
                                                        nullptr, B3, 512, 256,
                                                        256, 0);

  // 5) dominant pair GEMM + fused norm/relu/cls/log_softmax
  pair_kernel<<<8192, 256, 0, stream>>>(cau, emo, A3, B3, W3_W, cls_W, cls_b,
                                        sqne, sqnc, p);

  // 6) L_cau one-hot
  lcau_kernel<<<1024, 256, 0, stream>>>(label3, lcau);
}